// LongAttention_81028853006589
// MI455X (gfx1250) — compile-verified
//
#include <hip/hip_runtime.h>
#include <math.h>
#include <stdint.h>

// Problem dims (fixed by reference)
#define B_    2
#define T_    2048
#define DIM_  1024
#define HD_   8192
#define M1    (B_ * T_)      // 4096 rows (b,t)
#define N1    (4 * HD_)      // 32768  proj_in outputs
#define K1    DIM_           // 1024
#define K2    (2 * HD_)      // 16384  LN width / gemm2 K
#define N2    DIM_           // 1024   gemm2 N
#define NCH   16             // scan chunks along T
#define TC    (T_ / NCH)     // 128
#define LN_EPS 1e-5f
#define PI_F  3.14159265358979323846f

// GEMM tiling: 256 threads = 8 waves; wave grid 2(M) x 4(N); wave = 16x64 C strip
// Double-buffered LDS, BK=16, padded stride 20 (bank-conflict-free b64 frags,
// 16B-aligned rows). LDS = 2*(32+256)*20*4 = 46080 B.
#define BM 32
#define BN 256
#define BK 16
#define LDS_K (BK + 4)   // 20

static_assert(M1 % BM == 0 && N1 % BN == 0 && K1 % BK == 0, "tile div");
static_assert(K2 % BK == 0 && N2 % BN == 0, "tile div 2");

typedef __attribute__((ext_vector_type(2))) float v2f;
typedef __attribute__((ext_vector_type(8))) float v8f;

#if __has_builtin(__builtin_amdgcn_wmma_f32_16x16x4_f32)
#define USE_WMMA 1
#else
#define USE_WMMA 0
#endif

// Async global->LDS staging (ASYNCcnt-tracked) via inline CDNA5 asm.
// GV form: global_load_async_to_lds_b128 vdst(LDS byte addr), v[addr64], off
// The low 32 bits of a flat LDS pointer are the LDS byte offset (shared
// aperture occupies the high word), so truncation yields vdst.
#if defined(__gfx1250__)
#define USE_ASYNC 1
#else
#define USE_ASYNC 0
#endif

#if USE_ASYNC
__device__ __forceinline__ void async_ld_b128(const float* g, float* l) {
  asm volatile("global_load_async_to_lds_b128 %0, %1, off"
               :
               : "v"((uint32_t)(uintptr_t)l), "v"((uint64_t)(uintptr_t)g)
               : "memory");
}
#if __has_builtin(__builtin_amdgcn_s_wait_asynccnt)
#define WAIT_ASYNC(n) __builtin_amdgcn_s_wait_asynccnt(n)
#else
#define WAIT_ASYNC(n) asm volatile("s_wait_asynccnt %0" ::"i"(n) : "memory")
#endif
#endif

// ---------------------------------------------------------------------------
// Unified fp32 WMMA GEMM: out[m,n] = sum_k a'(m,k) * Bw[n,k]
//   DO_LN=false: a' = A[m,k]                         (GEMM1: raw = x @ W_in^T)
//   DO_LN=true : a' = (A[m,k]-mu_m)*rsig_m*g[k]+b[k] (GEMM2: LN fused on A)
// B panel staged with async global->LDS copies (ASYNCcnt), double buffered so
// the HBM fetch of tile k+1 overlaps the 16-WMMA compute phase of tile k.
// ---------------------------------------------------------------------------
template <int KTOT, int NOUT, bool DO_LN>
__global__ __launch_bounds__(256) void gemm_wmma_kernel(
    const float* __restrict__ A, const float* __restrict__ Bw,
    const float* __restrict__ gamma, const float* __restrict__ beta,
    const float2* __restrict__ stats, float* __restrict__ out) {
  __shared__ alignas(16) float As[2][BM * LDS_K];
  __shared__ alignas(16) float Bs[2][BN * LDS_K];

  const int tid   = threadIdx.x;
  const int lane  = tid & 31;
  const int wid   = tid >> 5;
  const int wm    = wid >> 2;      // 0..1
  const int wn    = wid & 3;       // 0..3
  const int lhalf = lane >> 4;     // f32 frag: 0 -> K+0/1, 1 -> K+2/3
  const int lrow  = lane & 15;
  const int m0 = (DO_LN ? blockIdx.y : blockIdx.x) * BM;
  const int n0 = (DO_LN ? blockIdx.x : blockIdx.y) * BN;

  const int ar = tid >> 3;         // A stage: 32 rows x 8 float2
  const int ac = (tid & 7) * 2;
  const int br = tid >> 2;         // B stage: 4x (64 rows x 4 float4)
  const int bc = (tid & 3) * 4;

  constexpr int NT = KTOT / BK;

  v8f c0 = {0,0,0,0,0,0,0,0}, c1 = {0,0,0,0,0,0,0,0};
  v8f c2 = {0,0,0,0,0,0,0,0}, c3 = {0,0,0,0,0,0,0,0};

  // ---- prologue: fetch tile 0 ----
  float2 st = make_float2(0.f, 1.f);
  float2 greg, breg;
  if constexpr (DO_LN) {
    st   = stats[m0 + ar];
    greg = *(const float2*)(gamma + ac);
    breg = *(const float2*)(beta + ac);
  }
  float2 areg = *(const float2*)(A + (size_t)(m0 + ar) * KTOT + ac);
#if USE_ASYNC
  #pragma unroll
  for (int it = 0; it < 4; ++it)
    async_ld_b128(Bw + (size_t)(n0 + br + it * 64) * KTOT + bc,
                  &Bs[0][(br + it * 64) * LDS_K + bc]);
#else
  float4 bregs[4];
  #pragma unroll
  for (int it = 0; it < 4; ++it)
    bregs[it] = *(const float4*)(Bw + (size_t)(n0 + br + it * 64) * KTOT + bc);
#endif

  for (int kb = 0; kb < NT; ++kb) {
    const int p = kb & 1;
    __syncthreads();   // all waves finished computing tile kb-1 (buffers free)
    {                  // commit A(kb) to LDS, LN affine fused in-register
      float2 v = areg;
      if constexpr (DO_LN) {
        v.x = (v.x - st.x) * st.y * greg.x + breg.x;
        v.y = (v.y - st.x) * st.y * greg.y + breg.y;
      }
      *(float2*)&As[p][ar * LDS_K + ac] = v;
    }
#if !USE_ASYNC
    #pragma unroll
    for (int it = 0; it < 4; ++it)
      *(float4*)&Bs[p][(br + it * 64) * LDS_K + bc] = bregs[it];
#endif
    if (kb + 1 < NT) {   // prefetch tile kb+1 while tile kb computes
      const int k0n = (kb + 1) * BK;
      areg = *(const float2*)(A + (size_t)(m0 + ar) * KTOT + k0n + ac);
      if constexpr (DO_LN) {
        greg = *(const float2*)(gamma + k0n + ac);
        breg = *(const float2*)(beta + k0n + ac);
      }
#if USE_ASYNC
      #pragma unroll
      for (int it = 0; it < 4; ++it)
        async_ld_b128(Bw + (size_t)(n0 + br + it * 64) * KTOT + k0n + bc,
                      &Bs[1 - p][(br + it * 64) * LDS_K + bc]);
      WAIT_ASYNC(4);     // tile kb's 4 async copies done; kb+1's may be in flight
#else
      #pragma unroll
      for (int it = 0; it < 4; ++it)
        bregs[it] = *(const float4*)(Bw + (size_t)(n0 + br + it * 64) * KTOT + k0n + bc);
#endif
    }
#if USE_ASYNC
    else {
      WAIT_ASYNC(0);
    }
#endif
    __syncthreads();   // tile kb staged by all waves and visible

    const float* as = &As[p][(wm * 16 + lrow) * LDS_K + lhalf * 2];
    const float* bs = &Bs[p][(wn * 64 + lrow) * LDS_K + lhalf * 2];
    #pragma unroll
    for (int kk = 0; kk < BK; kk += 4) {
#if USE_WMMA
      v2f a  = *(const v2f*)(as + kk);
      v2f b0 = *(const v2f*)(bs + kk);
      v2f b1 = *(const v2f*)(bs + 16 * LDS_K + kk);
      v2f b2 = *(const v2f*)(bs + 32 * LDS_K + kk);
      v2f b3 = *(const v2f*)(bs + 48 * LDS_K + kk);
      c0 = __builtin_amdgcn_wmma_f32_16x16x4_f32(false, a, false, b0, (short)0, c0, false, false);
      c1 = __builtin_amdgcn_wmma_f32_16x16x4_f32(false, a, false, b1, (short)0, c1, false, false);
      c2 = __builtin_amdgcn_wmma_f32_16x16x4_f32(false, a, false, b2, (short)0, c2, false, false);
      c3 = __builtin_amdgcn_wmma_f32_16x16x4_f32(false, a, false, b3, (short)0, c3, false, false);
#else
      #pragma unroll
      for (int kq = 0; kq < 4; ++kq) {
        #pragma unroll
        for (int v = 0; v < 8; ++v) {
          float av = As[p][(wm * 16 + v + 8 * lhalf) * LDS_K + kk + kq];
          c0[v] += av * Bs[p][(wn * 64 +  0 + lrow) * LDS_K + kk + kq];
          c1[v] += av * Bs[p][(wn * 64 + 16 + lrow) * LDS_K + kk + kq];
          c2[v] += av * Bs[p][(wn * 64 + 32 + lrow) * LDS_K + kk + kq];
          c3[v] += av * Bs[p][(wn * 64 + 48 + lrow) * LDS_K + kk + kq];
        }
      }
#endif
    }
  }

  #pragma unroll
  for (int v = 0; v < 8; ++v) {   // C layout: VGPR v = row v (lanes<16) / v+8
    int m = m0 + wm * 16 + v + 8 * lhalf;
    size_t rowb = (size_t)m * NOUT + n0 + wn * 64 + lrow;
    __builtin_nontemporal_store(c0[v], &out[rowb +  0]);
    __builtin_nontemporal_store(c1[v], &out[rowb + 16]);
    __builtin_nontemporal_store(c2[v], &out[rowb + 32]);
    __builtin_nontemporal_store(c3[v], &out[rowb + 48]);
  }
}

// ---------------------------------------------------------------------------
// Scan pass 1: per (b,h,chunk) partial sum of key phasor over TC timesteps.
// key = sigmoid(c_mag) * exp(i*(tanh(c_phase)*pi + pos_phase))
// raw is 512MB (never L2-resident) -> non-temporal loads protect pos/W reuse.
// ---------------------------------------------------------------------------
__global__ __launch_bounds__(256) void scan_partial_kernel(
    const float* __restrict__ raw, const float* __restrict__ pos,
    float2* __restrict__ csum) {
  const int h = blockIdx.x * 256 + threadIdx.x;
  const int chunk = blockIdx.y;
  const int b = blockIdx.z;
  float sre = 0.f, sim = 0.f;
  const int t0 = chunk * TC;
  for (int t = t0; t < t0 + TC; ++t) {
    size_t base = (size_t)(b * T_ + t) * N1;
    float cp = __builtin_nontemporal_load(raw + base + h);
    float cm = __builtin_nontemporal_load(raw + base + HD_ + h);
    float ang = tanhf(cp) * PI_F + pos[(size_t)t * HD_ + h];
    float mg = 1.f / (1.f + __expf(-cm));
    float sn, cs;
    __sincosf(ang, &sn, &cs);
    sre += mg * cs;
    sim += mg * sn;
  }
  csum[(size_t)(b * HD_ + h) * NCH + chunk] = make_float2(sre, sim);
}

// Scan pass 2: exclusive prefix over the NCH chunk sums of each channel.
__global__ __launch_bounds__(256) void scan_offsets_kernel(float2* __restrict__ csum) {
  const int ch = blockIdx.x * 256 + threadIdx.x;
  float2* p = &csum[(size_t)ch * NCH];
  float re = 0.f, im = 0.f;
  #pragma unroll
  for (int c = 0; c < NCH; ++c) {
    float2 v = p[c];
    p[c] = make_float2(re, im);
    re += v.x; im += v.y;
  }
}

// Scan pass 3: seeded inclusive scan + retrieval = cumsum * conj(query),
// interleaved (re, im) write -> ret[(b,t), 2*HD].
__global__ __launch_bounds__(256) void scan_apply_kernel(
    const float* __restrict__ raw, const float* __restrict__ pos,
    const float2* __restrict__ csum, float2* __restrict__ ret2) {
  const int h = blockIdx.x * 256 + threadIdx.x;
  const int chunk = blockIdx.y;
  const int b = blockIdx.z;
  float2 acc = csum[(size_t)(b * HD_ + h) * NCH + chunk];
  const int t0 = chunk * TC;
  for (int t = t0; t < t0 + TC; ++t) {
    size_t base = (size_t)(b * T_ + t) * N1;
    float cp = __builtin_nontemporal_load(raw + base + h);
    float cm = __builtin_nontemporal_load(raw + base + HD_ + h);
    float qr = __builtin_nontemporal_load(raw + base + 2 * HD_ + h);
    float qi = __builtin_nontemporal_load(raw + base + 3 * HD_ + h);
    float ang = tanhf(cp) * PI_F + pos[(size_t)t * HD_ + h];
    float mg = 1.f / (1.f + __expf(-cm));
    float sn, cs;
    __sincosf(ang, &sn, &cs);
    acc.x += mg * cs;
    acc.y += mg * sn;
    float rr = acc.x * qr + acc.y * qi;   // Re(m * conj(q))
    float ri = acc.y * qr - acc.x * qi;   // Im(m * conj(q))
    ret2[(size_t)(b * T_ + t) * HD_ + h] = make_float2(rr, ri);
  }
}

// ---------------------------------------------------------------------------
// LayerNorm row statistics: one workgroup per (b,t) row of 16384.
// ---------------------------------------------------------------------------
__global__ __launch_bounds__(256) void row_stats_kernel(
    const float* __restrict__ ret, float2* __restrict__ stats) {
  __shared__ float ssum[256];
  __shared__ float ssq[256];
  const int m = blockIdx.x;
  const float4* r = (const float4*)(ret + (size_t)m * K2);
  float s = 0.f, q = 0.f;
  for (int i = threadIdx.x; i < K2 / 4; i += 256) {
    float4 v = r[i];
    s += v.x + v.y + v.z + v.w;
    q += v.x * v.x + v.y * v.y + v.z * v.z + v.w * v.w;
  }
  ssum[threadIdx.x] = s;
  ssq[threadIdx.x]  = q;
  __syncthreads();
  for (int off = 128; off > 0; off >>= 1) {
    if (threadIdx.x < off) {
      ssum[threadIdx.x] += ssum[threadIdx.x + off];
      ssq[threadIdx.x]  += ssq[threadIdx.x + off];
    }
    __syncthreads();
  }
  if (threadIdx.x == 0) {
    float mu  = ssum[0] / (float)K2;
    float var = ssq[0] / (float)K2 - mu * mu;
    stats[m] = make_float2(mu, rsqrtf(var + LN_EPS));
  }
}

// ---------------------------------------------------------------------------
extern "C" void kernel_launch(void* const* d_in, const int* in_sizes, int n_in,
                              void* d_out, int out_size, void* d_ws, size_t ws_size,
                              hipStream_t stream) {
  (void)in_sizes; (void)n_in; (void)out_size; (void)ws_size;
  const float* x      = (const float*)d_in[0];
  const float* W_in   = (const float*)d_in[1];
  const float* W_out  = (const float*)d_in[2];
  const float* gamma  = (const float*)d_in[3];
  const float* beta   = (const float*)d_in[4];
  const float* pos    = (const float*)d_in[5];

  // Workspace layout (floats): raw [M1*N1] | ret [M1*K2] | csum | stats
  float*  raw   = (float*)d_ws;
  float*  ret   = raw + (size_t)M1 * N1;
  float2* csum  = (float2*)(ret + (size_t)M1 * K2);
  float2* stats = csum + (size_t)B_ * HD_ * NCH;

  // GEMM1: x-blocks = M (fast) so co-resident WGs share one W_in N-panel;
  // W_in (128MB) and x (16MB) both fit the 192MB L2.
  gemm_wmma_kernel<K1, N1, false><<<dim3(M1 / BM, N1 / BN), 256, 0, stream>>>(
      x, W_in, nullptr, nullptr, nullptr, raw);
  scan_partial_kernel<<<dim3(HD_ / 256, NCH, B_), 256, 0, stream>>>(raw, pos, csum);
  scan_offsets_kernel<<<dim3((B_ * HD_) / 256), 256, 0, stream>>>(csum);
  scan_apply_kernel<<<dim3(HD_ / 256, NCH, B_), 256, 0, stream>>>(raw, pos, csum, (float2*)ret);
  row_stats_kernel<<<dim3(M1), 256, 0, stream>>>(ret, stats);
  // GEMM2: x-blocks = N (4) so one 32-row ret block feeds all 4 N-panels;
  // W_out (67MB) stays L2-resident across the whole kernel.
  gemm_wmma_kernel<K2, N2, true><<<dim3(N2 / BN, M1 / BM), 256, 0, stream>>>(
      ret, W_out, gamma, beta, (const float2*)stats, (float*)d_out);
}